// GNNBlock_84963043049900
// MI455X (gfx1250) — compile-verified
//
#include <hip/hip_runtime.h>

// ---------------------------------------------------------------------------
// 2-layer GCN (GCNConv -> BN -> PReLU) x2 for MI455X (gfx1250, wave32).
// GEMMs use V_WMMA_F32_16X16X4_F32 (exact fp32), one wave = 16x64 output strip
// (4 accumulators, 4x A-fragment reuse). Edge aggregation is the bandwidth
// bottleneck (L2-resident gather + f32 atomics), done wave-per-edge.
// ---------------------------------------------------------------------------

#define NNODES   50000
#define D_IN     128
#define D_HID    256        // == D_OUT
#define CH       256        // channel count for both layers
#define BN_EPS   1e-5f

typedef __attribute__((ext_vector_type(2))) float v2f;
typedef __attribute__((ext_vector_type(8))) float v8f;

// ---------------- utility ----------------
__global__ void zero_f32_kernel(float* __restrict__ p, int n) {
    int i = blockIdx.x * blockDim.x + threadIdx.x;
    if (i < n) p[i] = 0.0f;
}

// ---------------- degree / norm ----------------
__global__ void deg_scatter_kernel(const int* __restrict__ dst, float* __restrict__ deg, int E) {
    int i = blockIdx.x * blockDim.x + threadIdx.x;
    if (i < E) atomicAdd(&deg[dst[i]], 1.0f);
}

// deg -> dinv = rsqrt(deg + 1)   (+1 is the self loop)
__global__ void deg_finalize_kernel(float* __restrict__ deg, int n) {
    int i = blockIdx.x * blockDim.x + threadIdx.x;
    if (i < n) deg[i] = rsqrtf(deg[i] + 1.0f);
}

// ---------------- fp32 WMMA GEMM:  C[M,256] = A[M,K] @ B[K,256] ----------------
// One wave computes a 16x64 output strip (4 accumulators), stepping K by 4 with
// v_wmma_f32_16x16x4_f32; each A fragment is reused for 4 WMMAs.
// A 128-thread block (4 waves) covers all 256 columns of one 16-row strip.
// M must be a multiple of 16 (50000 = 3125*16).
// Per-lane layouts (ISA 7.12.2):
//   A: (M = lane&15, K = k + 2*(lane>>4) + {0,1})
//   B: (N = lane&15, K = k + 2*(lane>>4) + {0,1})
//   C: VGPR v -> (M = v + 8*(lane>>4), N = lane&15)
__global__ void gemm_wmma_f32_kernel(const float* __restrict__ A,
                                     const float* __restrict__ B,
                                     float* __restrict__ C,
                                     int K) {
    const int lane  = threadIdx.x & 31;
    const int wave  = threadIdx.x >> 5;          // 0..3
    const int tileM = blockIdx.x;                // 0..3124
    const int n0    = wave * 64;                 // this wave: cols n0..n0+63
    const int half  = lane >> 4;                 // 0 or 1
    const int l     = lane & 15;

    const float* __restrict__ arow  = A + (size_t)(tileM * 16 + l) * K;
    const float* __restrict__ bbase = B + n0 + l;

    v8f acc0 = {}, acc1 = {}, acc2 = {}, acc3 = {};
    for (int k = 0; k < K; k += 4) {
        const int kk = k + 2 * half;
        v2f a;
        a.x = arow[kk];
        a.y = arow[kk + 1];
        const float* __restrict__ bk0 = bbase + (size_t)kk * CH;
        const float* __restrict__ bk1 = bbase + (size_t)(kk + 1) * CH;
        v2f b;
        b.x = bk0[0];  b.y = bk1[0];
        acc0 = __builtin_amdgcn_wmma_f32_16x16x4_f32(false, a, false, b, (short)0, acc0, false, false);
        b.x = bk0[16]; b.y = bk1[16];
        acc1 = __builtin_amdgcn_wmma_f32_16x16x4_f32(false, a, false, b, (short)0, acc1, false, false);
        b.x = bk0[32]; b.y = bk1[32];
        acc2 = __builtin_amdgcn_wmma_f32_16x16x4_f32(false, a, false, b, (short)0, acc2, false, false);
        b.x = bk0[48]; b.y = bk1[48];
        acc3 = __builtin_amdgcn_wmma_f32_16x16x4_f32(false, a, false, b, (short)0, acc3, false, false);
    }

    float* __restrict__ crow = C + (size_t)(tileM * 16 + half * 8) * CH + n0 + l;
#pragma unroll
    for (int v = 0; v < 8; ++v) {
        float* r = crow + (size_t)v * CH;
        r[0]  = acc0[v];
        r[16] = acc1[v];
        r[32] = acc2[v];
        r[48] = acc3[v];
    }
}

// ---------------- aggregation ----------------
// out[i][c] = dinv[i]^2 * H[i][c] + bias[c]   (self-loop term + conv bias)
__global__ void agg_init_kernel(const float* __restrict__ H,
                                const float* __restrict__ dinv,
                                const float* __restrict__ bias,
                                float* __restrict__ out, int n) {
    int i = blockIdx.x * blockDim.x + threadIdx.x;
    if (i < n) {
        int row = i >> 8;          // /256
        int c   = i & 255;
        float di = dinv[row];
        out[i] = di * di * H[i] + bias[c];
    }
}

// Wave-per-edge scatter: out[dst] += dinv[src]*dinv[dst] * H[src]
// Each lane: 2 x float4 gathers (B128) + 8 dword f32 atomics.
__global__ void edge_scatter_kernel(const float* __restrict__ H,
                                    const int* __restrict__ src,
                                    const int* __restrict__ dst,
                                    const float* __restrict__ dinv,
                                    float* __restrict__ out, int E) {
    const int edge = (int)((blockIdx.x * blockDim.x + threadIdx.x) >> 5);
    const int lane = threadIdx.x & 31;
    if (edge >= E) return;

    const int s = src[edge];
    const int d = dst[edge];
    const float norm = dinv[s] * dinv[d];

    const float4* __restrict__ hs = (const float4*)(H + (size_t)s * CH);
    float* __restrict__ ob = out + (size_t)d * CH;

#pragma unroll
    for (int r = 0; r < 2; ++r) {
        const int i = lane + r * 32;          // float4 index, covers 256 channels
        const float4 v = hs[i];
        float* o = ob + i * 4;
        atomicAdd(o + 0, norm * v.x);
        atomicAdd(o + 1, norm * v.y);
        atomicAdd(o + 2, norm * v.z);
        atomicAdd(o + 3, norm * v.w);
    }
}

// ---------------- BatchNorm (training mode) + PReLU ----------------
#define BN_ROWS_PER_BLOCK 64

__global__ void bn_stats_kernel(const float* __restrict__ h,
                                float* __restrict__ sum,
                                float* __restrict__ sumsq, int rows) {
    const int c  = threadIdx.x;                       // 256 threads = 256 channels
    const int r0 = blockIdx.x * BN_ROWS_PER_BLOCK;
    const int r1 = min(r0 + BN_ROWS_PER_BLOCK, rows);
    float s = 0.0f, q = 0.0f;
    for (int r = r0; r < r1; ++r) {
        float v = h[(size_t)r * CH + c];
        s += v;
        q += v * v;
    }
    atomicAdd(&sum[c], s);
    atomicAdd(&sumsq[c], q);
}

__global__ void bn_finalize_kernel(const float* __restrict__ sum,
                                   const float* __restrict__ sumsq,
                                   const float* __restrict__ w,
                                   const float* __restrict__ b,
                                   float* __restrict__ scale,
                                   float* __restrict__ shift) {
    const int c = threadIdx.x;
    const float inv_n = 1.0f / (float)NNODES;
    float mean = sum[c] * inv_n;
    float var  = sumsq[c] * inv_n - mean * mean;   // biased variance
    float sc   = w[c] * rsqrtf(var + BN_EPS);
    scale[c] = sc;
    shift[c] = b[c] - mean * sc;
}

__global__ void bn_prelu_apply_kernel(const float* __restrict__ h,
                                      const float* __restrict__ scale,
                                      const float* __restrict__ shift,
                                      const float* __restrict__ alpha,
                                      float* __restrict__ out, int n) {
    int i = blockIdx.x * blockDim.x + threadIdx.x;
    if (i < n) {
        int c = i & 255;
        float y = h[i] * scale[c] + shift[c];
        out[i] = (y >= 0.0f) ? y : alpha[0] * y;
    }
}

// ---------------------------------------------------------------------------
extern "C" void kernel_launch(void* const* d_in, const int* in_sizes, int n_in,
                              void* d_out, int out_size, void* d_ws, size_t ws_size,
                              hipStream_t stream) {
    // inputs per setup_inputs() order
    const float* x     = (const float*)d_in[0];
    const int*   eidx  = (const int*)  d_in[1];   // [2, E] row-major
    const float* W0    = (const float*)d_in[2];
    const float* b0    = (const float*)d_in[3];
    const float* bnw0  = (const float*)d_in[4];
    const float* bnb0  = (const float*)d_in[5];
    const float* a0    = (const float*)d_in[6];
    const float* W1    = (const float*)d_in[7];
    const float* b1    = (const float*)d_in[8];
    const float* bnw1  = (const float*)d_in[9];
    const float* bnb1  = (const float*)d_in[10];
    const float* a1    = (const float*)d_in[11];

    const int E = in_sizes[1] / 2;
    const int* src = eidx;        // edge_index[0]
    const int* dst = eidx + E;    // edge_index[1]

    // workspace layout
    float* dinv  = (float*)d_ws;                 // N        (deg, then rsqrt in place)
    float* H     = dinv + NNODES;                // N*256
    float* AGG   = H + (size_t)NNODES * CH;      // N*256
    float* sum   = AGG + (size_t)NNODES * CH;    // 256
    float* sumsq = sum + CH;                     // 256
    float* scale = sumsq + CH;                   // 256
    float* shift = scale + CH;                   // 256

    float* outp = (float*)d_out;

    const int elemsNC  = NNODES * CH;
    const int gemmGrid  = NNODES / 16;           // 3125 row-strips
    const int gemmBlock = 128;                   // 4 waves x 16x64 strip = 16x256
    const int scatterGrid = (E + 7) / 8;         // 8 edges (waves) per 256-thread block
    const int bnGrid = (NNODES + BN_ROWS_PER_BLOCK - 1) / BN_ROWS_PER_BLOCK;

    // ---- normalization coefficients ----
    zero_f32_kernel<<<(NNODES + 255) / 256, 256, 0, stream>>>(dinv, NNODES);
    deg_scatter_kernel<<<(E + 255) / 256, 256, 0, stream>>>(dst, dinv, E);
    deg_finalize_kernel<<<(NNODES + 255) / 256, 256, 0, stream>>>(dinv, NNODES);

    // ---- layer 1 ----
    gemm_wmma_f32_kernel<<<gemmGrid, gemmBlock, 0, stream>>>(x, W0, H, D_IN);
    agg_init_kernel<<<(elemsNC + 255) / 256, 256, 0, stream>>>(H, dinv, b0, AGG, elemsNC);
    edge_scatter_kernel<<<scatterGrid, 256, 0, stream>>>(H, src, dst, dinv, AGG, E);
    zero_f32_kernel<<<2, 256, 0, stream>>>(sum, 2 * CH);   // sum + sumsq contiguous
    bn_stats_kernel<<<bnGrid, CH, 0, stream>>>(AGG, sum, sumsq, NNODES);
    bn_finalize_kernel<<<1, CH, 0, stream>>>(sum, sumsq, bnw0, bnb0, scale, shift);
    bn_prelu_apply_kernel<<<(elemsNC + 255) / 256, 256, 0, stream>>>(AGG, scale, shift, a0, AGG, elemsNC);

    // ---- layer 2 ----
    gemm_wmma_f32_kernel<<<gemmGrid, gemmBlock, 0, stream>>>(AGG, W1, H, D_HID);
    agg_init_kernel<<<(elemsNC + 255) / 256, 256, 0, stream>>>(H, dinv, b1, AGG, elemsNC);
    edge_scatter_kernel<<<scatterGrid, 256, 0, stream>>>(H, src, dst, dinv, AGG, E);
    zero_f32_kernel<<<2, 256, 0, stream>>>(sum, 2 * CH);
    bn_stats_kernel<<<bnGrid, CH, 0, stream>>>(AGG, sum, sumsq, NNODES);
    bn_finalize_kernel<<<1, CH, 0, stream>>>(sum, sumsq, bnw1, bnb1, scale, shift);
    bn_prelu_apply_kernel<<<(elemsNC + 255) / 256, 256, 0, stream>>>(AGG, scale, shift, a1, outp, elemsNC);
}